// NeuralMemory_29755533427428
// MI455X (gfx1250) — compile-verified
//
#include <hip/hip_runtime.h>
#include <math.h>

// ---------------------------------------------------------------------------
// NeuralMemory on gfx1250 (MI455X, wave32, WMMA)
//
// B=8, S=1024, D=128.
// Math transform: outputs y depend only on diag(M) which evolves
// independently -> 1024 parallel scalar scans. Off-diagonal (and diagonal)
// final states are linear in the driving terms u_t = k_t[i]*err_t[j]:
//   M_final = K^T diag(-theta*cM) Err ,  S_final = K^T diag(-theta*cS) Err
// which we evaluate with V_WMMA_F32_16X16X4_F32 chains.
// ---------------------------------------------------------------------------

typedef __attribute__((ext_vector_type(2))) float v2f;
typedef __attribute__((ext_vector_type(8))) float v8f;

#define BATCH 8
#define SEQ   1024
#define DIM   128

__device__ inline v8f wmma4(v2f a, v2f b, v8f c) {
  // (neg_a, A, neg_b, B, c_mod, C, reuse_a, reuse_b)
  return __builtin_amdgcn_wmma_f32_16x16x4_f32(false, a, false, b, (short)0, c,
                                               false, false);
}

// ---------------------------------------------------------------------------
// Kernel 1: projections k = x@Wk, v = x@Wv, q = x@Wq via WMMA.
// Output viewed as [8192,128]; 3 mats * 512 row-tiles * 8 col-tiles = 12288
// 16x16 tiles, one tile per wave, K=128 in steps of 4.
// ---------------------------------------------------------------------------
__global__ void proj_kernel(const float* __restrict__ X,
                            const float* __restrict__ Wk,
                            const float* __restrict__ Wv,
                            const float* __restrict__ Wq,
                            float* __restrict__ Kb,
                            float* __restrict__ Vb,
                            float* __restrict__ Qb) {
  const int lane = threadIdx.x & 31;
  const int wave = threadIdx.x >> 5;
  const int tile = blockIdx.x * 8 + wave;        // 0..12287
  const int mat  = tile >> 12;                   // /4096
  const int rem  = tile & 4095;
  const int rowt = rem >> 3;
  const int colt = rem & 7;

  const float* __restrict__ W = (mat == 0) ? Wk : (mat == 1) ? Wv : Wq;
  float* __restrict__ O       = (mat == 0) ? Kb : (mat == 1) ? Vb : Qb;

  const int m    = lane & 15;                    // row (A) / col (B,C)
  const int koff = (lane >> 4) << 1;             // K sub-offset 0 or 2
  const int row0 = rowt << 4;
  const int col0 = colt << 4;

  const float* xrow = X + (size_t)(row0 + m) * DIM;
  v8f c = {};
  for (int kk = 0; kk < DIM; kk += 4) {
    const int kb = kk + koff;
    v2f a;  a.x = xrow[kb];                 a.y = xrow[kb + 1];
    v2f b;  b.x = W[kb * DIM + col0 + m];   b.y = W[(kb + 1) * DIM + col0 + m];
    c = wmma4(a, b, c);
  }
  const int orow = row0 + (lane >> 4) * 8;       // C/D layout: M=v (+8 hi half)
#pragma unroll
  for (int v = 0; v < 8; ++v)
    O[(size_t)(orow + v) * DIM + col0 + m] = c[v];
}

// ---------------------------------------------------------------------------
// Kernel 2: gates = sigmoid(x@Wg + bg), Wg: [128,3]. One row per thread.
// ---------------------------------------------------------------------------
__global__ void gates_kernel(const float* __restrict__ X,
                             const float* __restrict__ Wg,
                             const float* __restrict__ bg,
                             float* __restrict__ alpha,
                             float* __restrict__ eta,
                             float* __restrict__ theta) {
  const int r = blockIdx.x * blockDim.x + threadIdx.x;   // 0..8191
  const float* xr = X + (size_t)r * DIM;
  float z0 = bg[0], z1 = bg[1], z2 = bg[2];
  for (int d = 0; d < DIM; ++d) {
    const float xv = xr[d];
    z0 += xv * Wg[d * 3 + 0];
    z1 += xv * Wg[d * 3 + 1];
    z2 += xv * Wg[d * 3 + 2];
  }
  alpha[r] = 1.f / (1.f + __expf(-z0));
  eta[r]   = 1.f / (1.f + __expf(-z1));
  theta[r] = 1.f / (1.f + __expf(-z2));
}

// ---------------------------------------------------------------------------
// Kernel 3: diagonal scan. One (b,d) state per thread: 8 blocks x 128 threads.
// Produces y (first output chunk) and err trajectories.
// ---------------------------------------------------------------------------
__global__ void scan_kernel(const float* __restrict__ Kb,
                            const float* __restrict__ Vb,
                            const float* __restrict__ Qb,
                            const float* __restrict__ alpha,
                            const float* __restrict__ eta,
                            const float* __restrict__ theta,
                            float* __restrict__ Err,
                            float* __restrict__ Y) {
  const int b = blockIdx.x;
  const int d = threadIdx.x;
  const size_t base  = (size_t)b * SEQ * DIM + d;
  const size_t gbase = (size_t)b * SEQ;
  float m = 0.f, s = 0.f;
  for (int t = 0; t < SEQ; ++t) {
    const size_t idx = base + (size_t)t * DIM;
    const float kt = Kb[idx], vt = Vb[idx], qt = Qb[idx];
    const float at = alpha[gbase + t], et = eta[gbase + t],
                tt = theta[gbase + t];
    const float y = qt * m;            // uses M_prev diag (matches scan_fn)
    const float e = kt * m - vt;       // pred - v
    Y[idx]   = y;
    Err[idx] = e;
    s = et * s - tt * (kt * e);        // diag of grad = k[j]*err[j]
    m = (1.f - at) * m + s;
  }
}

// ---------------------------------------------------------------------------
// Kernel 4: backward coefficient scan over gates only.
// cS_r = prod_{s>r} eta_s ; cM_r = PA_r + eta_{r+1} cM_{r+1},
// PA_r = prod_{s>r} (1-alpha_s). Emits wM = -theta*cM, wS = -theta*cS.
// ---------------------------------------------------------------------------
__global__ void coeff_kernel(const float* __restrict__ alpha,
                             const float* __restrict__ eta,
                             const float* __restrict__ theta,
                             float* __restrict__ wM,
                             float* __restrict__ wS) {
  const int b = threadIdx.x;
  if (b >= BATCH) return;
  const size_t g = (size_t)b * SEQ;
  float cM = 1.f, cS = 1.f, pa = 1.f;
  for (int r = SEQ - 1; r >= 0; --r) {
    wM[g + r] = -theta[g + r] * cM;
    wS[g + r] = -theta[g + r] * cS;
    if (r > 0) {
      const float Er = eta[g + r];
      const float Ar = 1.f - alpha[g + r];
      pa = Ar * pa;
      cM = pa + Er * cM;
      cS = Er * cS;
    }
  }
}

// ---------------------------------------------------------------------------
// Kernel 5: M_final/S_final = K^T diag(w) Err, per batch, via WMMA.
// 8 batches * 64 tiles = 512 tiles, one per wave; both accumulators share A.
// K-dim = time (1024), chained 16x16x4 f32 WMMAs.
// ---------------------------------------------------------------------------
__global__ void final_kernel(const float* __restrict__ Kb,
                             const float* __restrict__ Err,
                             const float* __restrict__ wM,
                             const float* __restrict__ wS,
                             float* __restrict__ Mout,
                             float* __restrict__ Sout) {
  const int lane = threadIdx.x & 31;
  const int wave = threadIdx.x >> 5;
  const int tile = blockIdx.x * 8 + wave;        // 0..511
  const int b    = tile >> 6;
  const int rem  = tile & 63;
  const int i0   = (rem >> 3) << 4;
  const int j0   = (rem & 7) << 4;

  const int m    = lane & 15;
  const int koff = (lane >> 4) << 1;
  const size_t bb = (size_t)b * SEQ * DIM;
  const size_t gb = (size_t)b * SEQ;

  v8f cm = {}, cs = {};
  for (int t0 = 0; t0 < SEQ; t0 += 4) {
    const int t = t0 + koff;
    v2f a;
    a.x = Kb[bb + (size_t)t * DIM + i0 + m];        // A[m, k] = k_t[i0+m]
    a.y = Kb[bb + (size_t)(t + 1) * DIM + i0 + m];
    const float e0 = Err[bb + (size_t)t * DIM + j0 + m];
    const float e1 = Err[bb + (size_t)(t + 1) * DIM + j0 + m];
    v2f bm; bm.x = e0 * wM[gb + t]; bm.y = e1 * wM[gb + t + 1];
    v2f bs; bs.x = e0 * wS[gb + t]; bs.y = e1 * wS[gb + t + 1];
    cm = wmma4(a, bm, cm);
    cs = wmma4(a, bs, cs);
  }
  const size_t ob  = (size_t)b * DIM * DIM;
  const int orow = i0 + (lane >> 4) * 8;
#pragma unroll
  for (int v = 0; v < 8; ++v) {
    Mout[ob + (size_t)(orow + v) * DIM + j0 + m] = cm[v];
    Sout[ob + (size_t)(orow + v) * DIM + j0 + m] = cs[v];
  }
}

// ---------------------------------------------------------------------------
extern "C" void kernel_launch(void* const* d_in, const int* in_sizes, int n_in,
                              void* d_out, int out_size, void* d_ws,
                              size_t ws_size, hipStream_t stream) {
  const float* X  = (const float*)d_in[0];
  const float* Wk = (const float*)d_in[1];
  const float* Wv = (const float*)d_in[2];
  const float* Wq = (const float*)d_in[3];
  const float* Wg = (const float*)d_in[4];
  const float* bg = (const float*)d_in[5];

  const size_t BSD = (size_t)BATCH * SEQ * DIM;   // 1,048,576
  const size_t BDD = (size_t)BATCH * DIM * DIM;   //   131,072
  const size_t BS  = (size_t)BATCH * SEQ;         //     8,192

  float* out  = (float*)d_out;
  float* Y    = out;              // [B,S,D]
  float* Mout = out + BSD;        // [B,D,D]
  float* Sout = Mout + BDD;       // [B,D,D]

  float* ws = (float*)d_ws;       // ~16.9 MB of fp32 scratch
  float* Kb = ws;
  float* Vb = Kb + BSD;
  float* Qb = Vb + BSD;
  float* Er = Qb + BSD;
  float* al = Er + BSD;
  float* et = al + BS;
  float* th = et + BS;
  float* wM = th + BS;
  float* wS = wM + BS;

  proj_kernel <<<1536, 256, 0, stream>>>(X, Wk, Wv, Wq, Kb, Vb, Qb);
  gates_kernel<<<  32, 256, 0, stream>>>(X, Wg, bg, al, et, th);
  scan_kernel <<<BATCH, DIM, 0, stream>>>(Kb, Vb, Qb, al, et, th, Er, Y);
  coeff_kernel<<<   1,  32, 0, stream>>>(al, et, th, wM, wS);
  final_kernel<<<  64, 256, 0, stream>>>(Kb, Er, wM, wS, Mout, Sout);
}